// GNNEncoder_35905926594603
// MI455X (gfx1250) — compile-verified
//
#include <hip/hip_runtime.h>

typedef __attribute__((ext_vector_type(2))) float v2f;
typedef __attribute__((ext_vector_type(8))) float v8f;

#define NN   50000
#define NE   1600000
#define FIN  128
#define DD   64
#define LL   32
#define RR   4

// ---------------------------------------------------------------------------
// Generic fp32 WMMA GEMM:  out[nrows, NCOLS] = act(A[nrows,K] @ W[K,NCOLS] + bias)
// optional residual add (res has same [nrows, NCOLS] layout).
// One wave -> one 16x16 tile via V_WMMA_F32_16X16X4_F32, K stepped by 4.
// Wave32 layouts (CDNA5 ISA 7.12.2):
//   A 16x4 : lane L(0..15): v0=K0,v1=K1 ; lane L+16: v0=K2,v1=K3 ; M=L%16
//   B 4x16 : v holds row K=2*(lane>>4)+v, col N=lane%16
//   C/D    : VGPR v -> row M = v + 8*(lane>>4), col N = lane%16
// ---------------------------------------------------------------------------
template<int K, int NCOLS, bool RELU, bool RESIDUAL>
__global__ __launch_bounds__(256) void gemm_bias_act(
    const float* __restrict__ A,
    const float* __restrict__ W,
    const float* __restrict__ bias,
    const float* __restrict__ res,
    float* __restrict__ out,
    int nrows)
{
    constexpr int CT = NCOLS / 16;   // column tiles per block row
    constexpr int RT = 8 / CT;       // row tiles per block (8 waves / block)
    const int wave = threadIdx.x >> 5;
    const int lane = threadIdx.x & 31;
    const int ct = wave % CT;
    const int rt = wave / CT;
    const int row0 = blockIdx.x * (RT * 16) + rt * 16;
    if (row0 >= nrows) return;                       // wave-uniform exit
    const int n0 = ct * 16;
    const int m  = lane & 15;
    const int kh = lane >> 4;

    const int arow = (row0 + m < nrows) ? (row0 + m) : (nrows - 1);
    const float* ap = A + (size_t)arow * K + 2 * kh;
    const float* bp = W + (size_t)(2 * kh) * NCOLS + n0 + m;

    v8f c = {};
#pragma unroll
    for (int k0 = 0; k0 < K; k0 += 4) {
        v2f a, b;
        const float2 av = *(const float2*)(ap + k0);  // 8B aligned: k0%4==0, 2*kh even
        a[0] = av.x; a[1] = av.y;
        b[0] = bp[(size_t)(k0 + 0) * NCOLS];
        b[1] = bp[(size_t)(k0 + 1) * NCOLS];
        c = __builtin_amdgcn_wmma_f32_16x16x4_f32(
                /*neg_a=*/false, a, /*neg_b=*/false, b,
                /*c_mod=*/(short)0, c, /*reuse_a=*/false, /*reuse_b=*/false);
    }

    const int col = n0 + m;
    const float bv = bias[col];
#pragma unroll
    for (int v = 0; v < 8; ++v) {
        const int row = row0 + v + 8 * kh;
        if (row < nrows) {
            float val = c[v] + bv;
            if (RELU)     val = fmaxf(val, 0.0f);
            if (RESIDUAL) val += res[(size_t)row * NCOLS + col];
            out[(size_t)row * NCOLS + col] = val;
        }
    }
}

// ---------------------------------------------------------------------------
// agg[dst[e]][:] += message[src[e]][:]   (fp32 atomics, 16 lanes/edge x float4)
// ---------------------------------------------------------------------------
__global__ __launch_bounds__(256) void scatter_add_edges(
    const long long* __restrict__ src,
    const long long* __restrict__ dst,
    const float* __restrict__ msg,
    float* __restrict__ agg,
    int nedges)
{
    const long long t = (long long)blockIdx.x * blockDim.x + threadIdx.x;
    const long long e = t >> 4;
    if (e >= nedges) return;
    const int c4 = (int)(t & 15) * 4;
    const long long s = src[e];
    const long long d = dst[e];
    const float4 v = *(const float4*)(msg + (size_t)s * DD + c4);
    float* base = agg + (size_t)d * DD + c4;
    atomicAdd(base + 0, v.x);
    atomicAdd(base + 1, v.y);
    atomicAdd(base + 2, v.z);
    atomicAdd(base + 3, v.w);
}

__global__ __launch_bounds__(256) void zero_f32(float* __restrict__ p, size_t n)
{
    size_t i = (size_t)blockIdx.x * blockDim.x + threadIdx.x;
    if (i < n) p[i] = 0.0f;
}

extern "C" void kernel_launch(void* const* d_in, const int* in_sizes, int n_in,
                              void* d_out, int out_size, void* d_ws, size_t ws_size,
                              hipStream_t stream)
{
    const float*     x     = (const float*)d_in[0];
    const long long* ei    = (const long long*)d_in[1];
    const float*     W_in  = (const float*)d_in[2];
    const float*     b_in  = (const float*)d_in[3];
    const float*     W_msg = (const float*)d_in[4];
    const float*     b_msg = (const float*)d_in[5];
    const float*     W_upd = (const float*)d_in[6];
    const float*     b_upd = (const float*)d_in[7];
    const float*     W_mu  = (const float*)d_in[8];
    const float*     b_mu  = (const float*)d_in[9];
    const float*     W_lv  = (const float*)d_in[10];
    const float*     b_lv  = (const float*)d_in[11];

    float* state = (float*)d_ws;                     // [NN, DD]
    float* msg   = state + (size_t)NN * DD;          // [NN, DD]
    float* agg   = msg   + (size_t)NN * DD;          // [NN, DD]

    const long long* src = ei;            // edge_index[0]
    const long long* dst = ei + NE;       // edge_index[1]

    float* mu = (float*)d_out;                       // [NN, LL]
    float* lv = mu + (size_t)NN * LL;                // [NN, LL]

    const unsigned g64 = (NN + 31) / 32;   // 32 rows/block when NCOLS==64
    const unsigned g32 = (NN + 63) / 64;   // 64 rows/block when NCOLS==32

    // state = relu(x @ W_in + b_in)
    gemm_bias_act<FIN, DD, true, false><<<g64, 256, 0, stream>>>(
        x, W_in, b_in, nullptr, state, NN);

    for (int r = 0; r < RR; ++r) {
        // message = relu(state @ W_msg[r] + b_msg[r])
        gemm_bias_act<DD, DD, true, false><<<g64, 256, 0, stream>>>(
            state, W_msg + (size_t)r * DD * DD, b_msg + (size_t)r * DD,
            nullptr, msg, NN);

        // agg = 0
        {
            const size_t n = (size_t)NN * DD;
            zero_f32<<<(unsigned)((n + 255) / 256), 256, 0, stream>>>(agg, n);
        }

        // agg[dst] += message[src]
        {
            const long long threads = (long long)NE * 16;
            scatter_add_edges<<<(unsigned)((threads + 255) / 256), 256, 0, stream>>>(
                src, dst, msg, agg, NE);
        }

        // state = state + relu(agg @ W_upd[r] + b_upd[r])
        gemm_bias_act<DD, DD, true, true><<<g64, 256, 0, stream>>>(
            agg, W_upd + (size_t)r * DD * DD, b_upd + (size_t)r * DD,
            state, state, NN);
    }

    // heads
    gemm_bias_act<DD, LL, false, false><<<g32, 256, 0, stream>>>(
        state, W_mu, b_mu, nullptr, mu, NN);
    gemm_bias_act<DD, LL, false, false><<<g32, 256, 0, stream>>>(
        state, W_lv, b_lv, nullptr, lv, NN);
}